// ResonanceYaRNScalingRotaryEmbedding_67980742361244
// MI455X (gfx1250) — compile-verified
//
#include <hip/hip_runtime.h>
#include <hip/hip_bf16.h>

// YaRN resonance RoPE cos/sin table generation for MI455X (gfx1250, wave32).
//
// Output layout (matches reference tuple return, concatenated flat):
//   d_out[0 .. rows*128)              = cos  (rows = out_size / 256)
//   d_out[rows*128 .. 2*rows*128)     = sin
//
// Bandwidth-bound: 16 MiB of stores dominate; per-thread work is 4 sincos +
// 4 modulos. Each block: 16 rows x 16 threads/row x 4 freqs/thread.

#define HALF_DIM 64
#define FULL_DIM 128
#define ROWS_PER_BLOCK 16
#define BLOCK_THREADS 256

__global__ __launch_bounds__(BLOCK_THREADS)
void yarn_rope_table_kernel(const int* __restrict__ position_ids,
                            const float* __restrict__ r_inv_freq,
                            const float* __restrict__ r_wavelengths,
                            float* __restrict__ out,
                            int L,      // number of valid entries in position_ids
                            int rows)   // output rows (== L for this problem)
{
    __shared__ float s_inv[HALF_DIM];
    __shared__ int   s_wav[HALF_DIM];

    const int tid = threadIdx.x;

    // gfx1250 prefetch of the position window this block gathers from
    // (indices are always <= row index, so the block's rows are an upper bound).
    if (tid == 0) {
        __builtin_prefetch(position_ids + (size_t)blockIdx.x * ROWS_PER_BLOCK, 0, 3);
    }

    // Stage the tiny frequency tables into LDS once per block.
    if (tid < HALF_DIM) {
        s_inv[tid] = r_inv_freq[tid];
        s_wav[tid] = (int)r_wavelengths[tid];
    }
    __syncthreads();

    const float MSCALE = 1.1386294361119891f;  // (0.1*ln(4.0)+1.0)*1.0

    const int c   = tid & (ROWS_PER_BLOCK - 1);   // 0..15 : which 4-freq chunk
    const int r   = tid >> 4;                     // 0..15 : row within block
    const int row = blockIdx.x * ROWS_PER_BLOCK + r;
    if (row >= rows) return;

    // Position index for this row (rows == L in practice; stay safe otherwise).
    int l = row;
    if (l >= L) l %= L;

    const int i0 = c * 4;

    float4 cv, sv;
    float* cvf = reinterpret_cast<float*>(&cv);
    float* svf = reinterpret_cast<float*>(&sv);

#pragma unroll
    for (int k = 0; k < 4; ++k) {
        const int i = i0 + k;
        const int w = s_wav[i];
        // idx = l mod wavelength_i  (w >= 1 by construction: round(2*pi/inv_freq))
        const unsigned idx = (unsigned)l % (unsigned)w;
        // Faithful gather through position_ids (reference does jnp.take).
        const float pos = (float)position_ids[idx];
        const float ang = pos * s_inv[i];          // guaranteed in [0, 2*pi)
        float s, cc;
        __sincosf(ang, &s, &cc);                   // v_sin_f32 + v_cos_f32
        cvf[k] = cc * MSCALE;
        svf[k] = s * MSCALE;
    }

    // cos block then sin block; each value stored to both halves of dim=128.
    float* __restrict__ cp = out + (size_t)row * FULL_DIM;
    float* __restrict__ sp = out + (size_t)rows * FULL_DIM + (size_t)row * FULL_DIM;

    *reinterpret_cast<float4*>(cp + i0)            = cv;   // global_store_b128
    *reinterpret_cast<float4*>(cp + HALF_DIM + i0) = cv;
    *reinterpret_cast<float4*>(sp + i0)            = sv;
    *reinterpret_cast<float4*>(sp + HALF_DIM + i0) = sv;
}

extern "C" void kernel_launch(void* const* d_in, const int* in_sizes, int n_in,
                              void* d_out, int out_size, void* d_ws, size_t ws_size,
                              hipStream_t stream) {
    // setup_inputs() order: x (unused, dtype carrier), position_ids,
    // r_inv_freq, r_wavelengths.
    const int*   position_ids  = (const int*)d_in[1];
    const float* r_inv_freq    = (const float*)d_in[2];
    const float* r_wavelengths = (const float*)d_in[3];
    float*       out           = (float*)d_out;

    const int L    = in_sizes[1];                 // 16384
    const int rows = out_size / (2 * FULL_DIM);   // cos+sin, 128 wide each

    const int blocks = (rows + ROWS_PER_BLOCK - 1) / ROWS_PER_BLOCK;
    yarn_rope_table_kernel<<<blocks, BLOCK_THREADS, 0, stream>>>(
        position_ids, r_inv_freq, r_wavelengths, out, L, rows);
}